// Renderer_18726057411150
// MI455X (gfx1250) — compile-verified
//
#include <hip/hip_runtime.h>
#include <math.h>

typedef __attribute__((ext_vector_type(2))) float v2f;
typedef __attribute__((ext_vector_type(8))) float v8f;

#define EPS_NRM 1e-8f
#define EPS_DIV 1e-15f

// ---------------------------------------------------------------------------
// Kernel 1: p_cam = R * (x - c) for all points, via V_WMMA_F32_16X16X4_F32.
// One wave handles 16 points: A(16x4) = (x-c) rows, B(4x16) = R^T columns,
// D(16x16) f32, of which N=0..2 hold the transformed coordinates.
// A layout (32-bit, 16x4):  lane m=L&15 holds row m; VGPR0 = K(hi?2:0),
//                           VGPR1 = K(hi?3:1), hi = L>>4.
// B layout (32-bit, 4x16):  lane n=L&15 holds col n; VGPR0 = K(hi?2:0),
//                           VGPR1 = K(hi?3:1).
// D layout: VGPR r -> M = r + 8*hi, N = L&15.
// pcam is padded to Ppad (multiple of 128) points so D stores need no bounds
// check; pre-WMMA path is branch-free (EXEC all-ones per ISA requirement).
// ---------------------------------------------------------------------------
__global__ void __launch_bounds__(256)
k_transform_wmma(const float* __restrict__ pts,   // (B,P,3)
                 const float* __restrict__ rot,   // (B,3,3)
                 const float* __restrict__ pos,   // (B,3)
                 float* __restrict__ pcam,        // (B,Ppad,3)
                 int P, int Ppad)
{
    const int b    = blockIdx.y;
    const int lane = threadIdx.x & 31;
    const int wave = threadIdx.x >> 5;
    const int n    = lane & 15;
    const int hi   = lane >> 4;
    const int base = blockIdx.x * 128 + wave * 16;   // 8 waves * 16 points

    const float cx = pos[b * 3 + 0];
    const float cy = pos[b * 3 + 1];
    const float cz = pos[b * 3 + 2];

    int pi = base + n;
    if (pi >= P) pi = P - 1;                         // clamp read, EXEC stays full
    const float* pp = pts + ((size_t)b * P + pi) * 3;
    const float px = pp[0] - cx;
    const float py = pp[1] - cy;
    const float pz = pp[2] - cz;

    v2f a;
    a.x = hi ? pz : px;       // K = 2 : 0
    a.y = hi ? 0.0f : py;     // K = 3 : 1

    // B[k][n] = rot[n][k]; clamp n and zero arithmetically -> no branch
    const int   nc   = n < 2 ? n : 2;
    const float nsel = (n < 3) ? 1.0f : 0.0f;
    v2f bm;
    bm.x = rot[b * 9 + nc * 3 + (hi ? 2 : 0)] * nsel;
    bm.y = hi ? 0.0f : rot[b * 9 + nc * 3 + 1] * nsel;

    v8f c = {};
    v8f d = __builtin_amdgcn_wmma_f32_16x16x4_f32(
        /*neg_a=*/false, a, /*neg_b=*/false, bm,
        /*c_mod=*/(short)0, c, /*reuse_a=*/false, /*reuse_b=*/false);

    if (n < 3) {              // single uniform-per-lane predicate, stores
#pragma unroll                // always in-bounds thanks to Ppad padding
        for (int r = 0; r < 8; ++r) {
            const int po = base + r + (hi ? 8 : 0);
            pcam[((size_t)b * Ppad + po) * 3 + n] = d[r];
        }
    }
}

// ---------------------------------------------------------------------------
// Kernel 2: per-face. Gathers camera-space vertices, computes the (un-
// normalized) cross-product normal, emits the normalized normal output and a
// packed 28-float (112B, 16B-aligned) feature record:
//   per vertex k: [nx ny nz  -pk.x -pk.y -pk.z  uk vk 1] ; +1 pad float.
// ---------------------------------------------------------------------------
__global__ void __launch_bounds__(256)
k_faces(const int*   __restrict__ faces,      // (F,3)
        const float* __restrict__ pcam,       // (B,Ppad,3)
        const float* __restrict__ uv,         // (B,P,2)
        float*       __restrict__ normal_out, // (B,F,3)
        float*       __restrict__ feat,       // (B,F,28)
        int P, int Ppad, int F, int BF)
{
    const int tid = blockIdx.x * blockDim.x + threadIdx.x;
    if (tid >= BF) return;
    const int b = tid / F;
    const int f = tid - b * F;

    const int i0 = faces[f * 3 + 0];
    const int i1 = faces[f * 3 + 1];
    const int i2 = faces[f * 3 + 2];

    const float* q0 = pcam + ((size_t)b * Ppad + i0) * 3;
    const float* q1 = pcam + ((size_t)b * Ppad + i1) * 3;
    const float* q2 = pcam + ((size_t)b * Ppad + i2) * 3;
    const float p0x = q0[0], p0y = q0[1], p0z = q0[2];
    const float p1x = q1[0], p1y = q1[1], p1z = q1[2];
    const float p2x = q2[0], p2y = q2[1], p2z = q2[2];

    const float e1x = p1x - p0x, e1y = p1y - p0y, e1z = p1z - p0z;
    const float e2x = p2x - p0x, e2y = p2y - p0y, e2z = p2z - p0z;
    const float nx = e1y * e2z - e1z * e2y;
    const float ny = e1z * e2x - e1x * e2z;
    const float nz = e1x * e2y - e1y * e2x;

    const float nlen = sqrtf(nx * nx + ny * ny + nz * nz + EPS_NRM) + EPS_DIV;
    normal_out[(size_t)tid * 3 + 0] = nx / nlen;
    normal_out[(size_t)tid * 3 + 1] = ny / nlen;
    normal_out[(size_t)tid * 3 + 2] = nz / nlen;

    const float u0 = uv[((size_t)b * P + i0) * 2 + 0];
    const float v0 = uv[((size_t)b * P + i0) * 2 + 1];
    const float u1 = uv[((size_t)b * P + i1) * 2 + 0];
    const float v1 = uv[((size_t)b * P + i1) * 2 + 1];
    const float u2 = uv[((size_t)b * P + i2) * 2 + 0];
    const float v2 = uv[((size_t)b * P + i2) * 2 + 1];

    float r[28];
    r[0]  = nx;  r[1]  = ny;  r[2]  = nz;
    r[3]  = -p0x; r[4] = -p0y; r[5] = -p0z;
    r[6]  = u0;  r[7]  = v0;  r[8]  = 1.0f;
    r[9]  = nx;  r[10] = ny;  r[11] = nz;
    r[12] = -p1x; r[13] = -p1y; r[14] = -p1z;
    r[15] = u1;  r[16] = v1;  r[17] = 1.0f;
    r[18] = nx;  r[19] = ny;  r[20] = nz;
    r[21] = -p2x; r[22] = -p2y; r[23] = -p2z;
    r[24] = u2;  r[25] = v2;  r[26] = 1.0f;
    r[27] = 0.0f;

    float4* dst = (float4*)(feat + (size_t)tid * 28);
    const float4* src = (const float4*)r;
#pragma unroll
    for (int j = 0; j < 7; ++j)
        dst[j] = src[j];   // 7 x global_store_b128
}

// ---------------------------------------------------------------------------
// Kernel 3: per-pixel Phong shading. Gathers one 112B face record (7 x b128),
// barycentric-interpolates 9 features, normalizes normal/eye, nearest-sample
// texture with wrap (ix = floor(frac(u)*TW), iy = floor((1-frac(v))*TH)),
// applies Phong, writes imrender and the improb passthrough coalesced.
// ---------------------------------------------------------------------------
__global__ void __launch_bounds__(256)
k_pixels(const int*   __restrict__ pixface,   // (B,HW)
         const float* __restrict__ pixw,      // (B,HW,3)
         const float* __restrict__ improb,    // (B,HW)
         const float* __restrict__ feat,      // (B,F,28)
         const float* __restrict__ tex,       // (B,3,TH,TW)
         const float* __restrict__ light,     // (B,3)
         const float* __restrict__ material,  // (B,3,3)
         const float* __restrict__ shin,      // (B,1)
         float*       __restrict__ imrender,  // (B,HW,3)
         float*       __restrict__ improb_out,// (B,HW)
         int HW, int F, int TH, int TW, int total)
{
    const int p = blockIdx.x * blockDim.x + threadIdx.x;
    if (p >= total) return;
    const int b = p / HW;

    const int   face = pixface[p];
    const float w0 = pixw[(size_t)p * 3 + 0];
    const float w1 = pixw[(size_t)p * 3 + 1];
    const float w2 = pixw[(size_t)p * 3 + 2];

    float fr[28];
    {
        const float4* rec = (const float4*)(feat + ((size_t)b * F + face) * 28);
        float4* frv = (float4*)fr;
#pragma unroll
        for (int j = 0; j < 7; ++j)
            frv[j] = rec[j];   // 7 x global_load_b128 (L2-resident buffer)
    }

    float imf[9];
#pragma unroll
    for (int d = 0; d < 9; ++d)
        imf[d] = w0 * fr[d] + w1 * fr[9 + d] + w2 * fr[18 + d];

    // normalize interpolated normal
    float nx = imf[0], ny = imf[1], nz = imf[2];
    float nl = sqrtf(nx * nx + ny * ny + nz * nz + EPS_NRM) + EPS_DIV;
    nx /= nl; ny /= nl; nz /= nl;
    // normalize interpolated eye direction
    float ex = imf[3], ey = imf[4], ez = imf[5];
    float el = sqrtf(ex * ex + ey * ey + ez * ez + EPS_NRM) + EPS_DIV;
    ex /= el; ey /= el; ez /= el;
    // normalized light (per batch, L0/L2 cached)
    float lx = light[b * 3 + 0], ly = light[b * 3 + 1], lz = light[b * 3 + 2];
    float ll = sqrtf(lx * lx + ly * ly + lz * lz + EPS_NRM) + EPS_DIV;
    lx /= ll; ly /= ll; lz /= ll;

    float cosT = nx * lx + ny * ly + nz * lz;
    cosT = fminf(fmaxf(cosT, 0.0f), 1.0f);
    const float rx = -lx + 2.0f * cosT * nx;
    const float ry = -ly + 2.0f * cosT * ny;
    const float rz = -lz + 2.0f * cosT * nz;
    float cosA = rx * ex + ry * ey + rz * ez;
    cosA = fminf(fmaxf(cosA, 1e-5f), 1.0f);
    cosA = powf(cosA, shin[b]);

    // nearest texture sample with wrap + y flip (algebraically reduced from
    // the reference grid_sample_nearest on tc = frac(uv)*2-1, tc.y *= -1)
    const float u  = imf[6];
    const float v  = imf[7];
    const float fu = u - floorf(u);
    const float fv = v - floorf(v);
    const int ix = (int)floorf(fu * (float)TW);
    const int iy = (int)floorf((1.0f - fv) * (float)TH);
    const float valid =
        (ix >= 0 && ix < TW && iy >= 0 && iy < TH) ? 1.0f : 0.0f;
    int ixc = ix < 0 ? 0 : (ix > TW - 1 ? TW - 1 : ix);
    int iyc = iy < 0 ? 0 : (iy > TH - 1 ? TH - 1 : iy);
    const size_t chs = (size_t)TH * TW;
    const float* tb = tex + (size_t)b * 3 * chs + (size_t)iyc * TW + ixc;
    const float t0 = tb[0]       * valid;
    const float t1 = tb[chs]     * valid;
    const float t2 = tb[2 * chs] * valid;

    const float pr   = improb[p];
    const float mask = imf[8];
    const float w    = mask * pr;

    const float* mat = material + b * 9;   // rows: amb, dif, spe
    float c0 = ((mat[0] + mat[3] * cosT) * t0 + mat[6] * cosA) * w;
    float c1 = ((mat[1] + mat[4] * cosT) * t1 + mat[7] * cosA) * w;
    float c2 = ((mat[2] + mat[5] * cosT) * t2 + mat[8] * cosA) * w;
    c0 = fminf(fmaxf(c0, 0.0f), 1.0f);
    c1 = fminf(fmaxf(c1, 0.0f), 1.0f);
    c2 = fminf(fmaxf(c2, 0.0f), 1.0f);

    imrender[(size_t)p * 3 + 0] = c0;
    imrender[(size_t)p * 3 + 1] = c1;
    imrender[(size_t)p * 3 + 2] = c2;
    improb_out[p] = pr;
}

// ---------------------------------------------------------------------------
extern "C" void kernel_launch(void* const* d_in, const int* in_sizes, int n_in,
                              void* d_out, int out_size, void* d_ws, size_t ws_size,
                              hipStream_t stream)
{
    const float* points  = (const float*)d_in[0];   // (B,P,3)
    const int*   faces   = (const int*)  d_in[1];   // (F,3)
    const float* rot     = (const float*)d_in[2];   // (B,3,3)
    const float* pos     = (const float*)d_in[3];   // (B,3)
    // d_in[4] camera_proj: only feeds points2d, which is dead w.r.t. outputs
    const float* uv      = (const float*)d_in[5];   // (B,P,2)
    const float* tex     = (const float*)d_in[6];   // (B,3,TH,TW)
    const float* light   = (const float*)d_in[7];   // (B,3)
    const float* mater   = (const float*)d_in[8];   // (B,3,3)
    const float* shin    = (const float*)d_in[9];   // (B,1)
    const int*   pixface = (const int*)  d_in[10];  // (B,H,W)
    const float* pixw    = (const float*)d_in[11];  // (B,H,W,3)
    const float* improb  = (const float*)d_in[12];  // (B,H,W,1)

    const int B  = in_sizes[3] / 3;
    const int P  = in_sizes[0] / (3 * B);
    const int F  = in_sizes[1] / 3;
    const int HW = in_sizes[10] / B;
    const long texHW = (long)(in_sizes[6] / (3 * B));
    int TH = (int)(sqrt((double)texHW) + 0.5);
    if (TH < 1) TH = 1;
    const int TW = (int)(texHW / TH);

    const int Ppad = (P + 127) & ~127;   // pad so WMMA D-stores never go OOB

    // workspace layout: [pcam (B*Ppad*3 f32)] [feat (B*F*28 f32, 16B aligned)]
    float* pcam = (float*)d_ws;
    size_t pcamBytes = (((size_t)B * Ppad * 3 * sizeof(float)) + 15) & ~(size_t)15;
    float* feat = (float*)((char*)d_ws + pcamBytes);

    // output layout: imrender (B*HW*3) | improb (B*HW) | normal1 (B*F*3)
    float* out        = (float*)d_out;
    float* imrender   = out;
    float* improb_out = out + (size_t)B * HW * 3;
    float* normal_out = improb_out + (size_t)B * HW;

    {   // K1: WMMA point transform
        dim3 grid(Ppad / 128, B);
        k_transform_wmma<<<grid, 256, 0, stream>>>(points, rot, pos, pcam, P, Ppad);
    }
    {   // K2: per-face normals + packed feature records
        const int BF = B * F;
        k_faces<<<(BF + 255) / 256, 256, 0, stream>>>(
            faces, pcam, uv, normal_out, feat, P, Ppad, F, BF);
    }
    {   // K3: per-pixel shading
        const int total = B * HW;
        k_pixels<<<(total + 255) / 256, 256, 0, stream>>>(
            pixface, pixw, improb, feat, tex, light, mater, shin,
            imrender, improb_out, HW, F, TH, TW, total);
    }
}